// GATEdgeWeightPredictor_58102317580449
// MI455X (gfx1250) — compile-verified
//
#include <hip/hip_runtime.h>
#include <hip/hip_bf16.h>
#include <stdint.h>

typedef float v2f __attribute__((ext_vector_type(2)));
typedef float v8f __attribute__((ext_vector_type(8)));

// ---------- helpers ----------
__device__ __forceinline__ unsigned int f32_ord(float f) {
  unsigned int u = __float_as_uint(f);
  return (u & 0x80000000u) ? ~u : (u | 0x80000000u);
}
__device__ __forceinline__ float ord_f32(unsigned int u) {
  return __uint_as_float((u & 0x80000000u) ? (u & 0x7fffffffu) : ~u);
}

__global__ void fill_zero_f(float* __restrict__ p, size_t n) {
  size_t i = (size_t)blockIdx.x * blockDim.x + threadIdx.x;
  if (i < n) p[i] = 0.f;
}

// ---------- fp32 WMMA GEMM: C[M,N] = A[M,K] @ B[K,N] ----------
// Requirements: K % 64 == 0, N % 16 == 0. Block = 256 threads (8 waves),
// each block computes a 128x16 output tile (wave w -> rows [w*16, w*16+16)).
__global__ __launch_bounds__(256)
void gemm_wmma_f32(const float* __restrict__ A, const float* __restrict__ B,
                   float* __restrict__ C, int M, int K, int N) {
  __shared__ float As[128][68];  // pad 68 -> conflict-free ds_read_b64, 16B rows
  __shared__ float Bs[16][68];   // Bs[n][k] (transposed W tile)
  const int tid = threadIdx.x;
  const int lane = tid & 31;
  const int wave = tid >> 5;
  const int m0 = blockIdx.y * 128;
  const int n0 = blockIdx.x * 16;
  const int mloc = lane & 15;
  const int khalf = (lane >> 4) << 1;  // 0 or 2 (ISA 16x4 A-layout)

  v8f acc = {0.f, 0.f, 0.f, 0.f, 0.f, 0.f, 0.f, 0.f};

  for (int k0 = 0; k0 < K; k0 += 64) {
    // stage A tile 128x64 (coalesced float4)
#pragma unroll
    for (int i = 0; i < 8; ++i) {
      int chunk = tid + 256 * i;
      int r = chunk >> 4;
      int c4 = (chunk & 15) << 2;
      int gr = m0 + r;
      float4 v = make_float4(0.f, 0.f, 0.f, 0.f);
      if (gr < M) v = *(const float4*)(A + (size_t)gr * K + k0 + c4);
      *(float4*)&As[r][c4] = v;
    }
    // stage B tile 64x16, transposed into Bs[n][k]
    {
      int n = tid & 15;
      int kb = tid >> 4;
#pragma unroll
      for (int i = 0; i < 4; ++i) {
        int k = kb + 16 * i;
        Bs[n][k] = B[(size_t)(k0 + k) * N + n0 + n];
      }
    }
    __syncthreads();
#pragma unroll
    for (int kk = 0; kk < 64; kk += 4) {
      v2f a = *(const v2f*)&As[wave * 16 + mloc][kk + khalf];
      v2f b = *(const v2f*)&Bs[mloc][kk + khalf];
      acc = __builtin_amdgcn_wmma_f32_16x16x4_f32(false, a, false, b,
                                                  (short)0, acc, false, false);
    }
    __syncthreads();
  }
  // D layout: VGPR j holds row j (lanes 0-15) / row j+8 (lanes 16-31)
  const int rbase = m0 + wave * 16 + ((lane >> 4) << 3);
  const int cn = n0 + (lane & 15);
#pragma unroll
  for (int j = 0; j < 8; ++j) {
    int r = rbase + j;
    if (r < M) C[(size_t)r * N + cn] = acc[j];
  }
}

// ---------- alpha_s / alpha_d: one wave per (node, head), C == 128 ----------
__global__ void alpha_dots(const float* __restrict__ xp,
                           const float* __restrict__ a_src,
                           const float* __restrict__ a_dst,
                           float* __restrict__ as_out, float* __restrict__ ad_out,
                           int Nn, int H, int C) {
  int gw = (int)((blockIdx.x * blockDim.x + threadIdx.x) >> 5);
  int lane = threadIdx.x & 31;
  if (gw >= Nn * H) return;
  int n = gw / H, h = gw - n * H;
  const float* xr = xp + (size_t)n * H * C + (size_t)h * C;
  const float* asr = a_src + (size_t)h * C;
  const float* adr = a_dst + (size_t)h * C;
  float ss = 0.f, sd = 0.f;
  for (int c = lane; c < C; c += 32) {
    float xv = xr[c];
    ss = fmaf(xv, asr[c], ss);
    sd = fmaf(xv, adr[c], sd);
  }
#pragma unroll
  for (int off = 16; off > 0; off >>= 1) {
    ss += __shfl_down(ss, off, 32);
    sd += __shfl_down(sd, off, 32);
  }
  if (lane == 0) { as_out[gw] = ss; ad_out[gw] = sd; }
}

// ---------- edge pass 1: e = leakyrelu(as[src]+ad[dst]); segment max ----------
__global__ void edge_scores_max(const int* __restrict__ row, const int* __restrict__ col,
                                const float* __restrict__ als, const float* __restrict__ ald,
                                float* __restrict__ ebuf, unsigned int* __restrict__ segmax,
                                int E, int Nn, int H) {
  int e = blockIdx.x * blockDim.x + threadIdx.x;
  if (e >= E + Nn) return;
  int s = (e < E) ? row[e] : (e - E);  // self loops appended
  int d = (e < E) ? col[e] : (e - E);
  for (int h = 0; h < H; ++h) {
    float v = als[s * H + h] + ald[d * H + h];
    v = (v > 0.f) ? v : v * 0.2f;
    ebuf[(size_t)e * H + h] = v;
    atomicMax(&segmax[d * H + h], f32_ord(v));
  }
}

// ---------- edge pass 2: p = exp(e - max); segment sum ----------
__global__ void edge_exp_sum(const int* __restrict__ row, const int* __restrict__ col,
                             float* __restrict__ ebuf, const unsigned int* __restrict__ segmax,
                             float* __restrict__ denom, int E, int Nn, int H) {
  int e = blockIdx.x * blockDim.x + threadIdx.x;
  if (e >= E + Nn) return;
  int d = (e < E) ? col[e] : (e - E);
  for (int h = 0; h < H; ++h) {
    float mx = ord_f32(segmax[d * H + h]);
    float p = expf(ebuf[(size_t)e * H + h] - mx);
    ebuf[(size_t)e * H + h] = p;
    atomicAdd(&denom[d * H + h], p);
  }
}

// ---------- edge pass 3: agg[dst] += alpha * xp[src] ----------
__global__ void edge_aggregate(const int* __restrict__ row, const int* __restrict__ col,
                               const float* __restrict__ ebuf, const float* __restrict__ denom,
                               const float* __restrict__ xp, float* __restrict__ agg,
                               int E, int Nn, int H, int C) {
  int e = blockIdx.x;
  int s = (e < E) ? row[e] : (e - E);
  int d = (e < E) ? col[e] : (e - E);
  int Cout = H * C;
  for (int c = threadIdx.x; c < Cout; c += blockDim.x) {
    int h = c / C;
    float alpha = ebuf[(size_t)e * H + h] / (denom[d * H + h] + 1e-16f);
    atomicAdd(&agg[(size_t)d * Cout + c], alpha * xp[(size_t)s * Cout + c]);
  }
}

// ---------- finalize: t = agg + bias; y = LN(t)*g+b; out = elu(y [+ resid]) ----------
__global__ __launch_bounds__(256)
void finalize_ln_elu(const float* __restrict__ agg, const float* __restrict__ bias,
                     const float* __restrict__ g, const float* __restrict__ b,
                     const float* __restrict__ resid, float* __restrict__ out, int Cout) {
  __shared__ float sred[256];
  int n = blockIdx.x;
  int tid = threadIdx.x;
  int c0 = tid, c1 = tid + 256;
  bool h0 = c0 < Cout, h1 = c1 < Cout;
  float v0 = h0 ? agg[(size_t)n * Cout + c0] + bias[c0] : 0.f;
  float v1 = h1 ? agg[(size_t)n * Cout + c1] + bias[c1] : 0.f;
  sred[tid] = v0 + v1;
  __syncthreads();
  for (int s = 128; s > 0; s >>= 1) {
    if (tid < s) sred[tid] += sred[tid + s];
    __syncthreads();
  }
  float mu = sred[0] / (float)Cout;
  __syncthreads();
  float d0 = h0 ? v0 - mu : 0.f;
  float d1 = h1 ? v1 - mu : 0.f;
  sred[tid] = d0 * d0 + d1 * d1;
  __syncthreads();
  for (int s = 128; s > 0; s >>= 1) {
    if (tid < s) sred[tid] += sred[tid + s];
    __syncthreads();
  }
  float inv = rsqrtf(sred[0] / (float)Cout + 1e-5f);
  if (h0) {
    float y = d0 * inv * g[c0] + b[c0];
    if (resid) y += resid[(size_t)n * Cout + c0];
    out[(size_t)n * Cout + c0] = (y > 0.f) ? y : expm1f(y);
  }
  if (h1) {
    float y = d1 * inv * g[c1] + b[c1];
    if (resid) y += resid[(size_t)n * Cout + c1];
    out[(size_t)n * Cout + c1] = (y > 0.f) ? y : expm1f(y);
  }
}

// ---------- edge predictor init: ew = b2 ----------
__global__ void init_ew(float* __restrict__ ew, const float* __restrict__ b2, int E) {
  int e = blockIdx.x * blockDim.x + threadIdx.x;
  if (e < E) ew[e] = b2[0];
}

// ---------- fused edge predictor: ew += relu(concat(h[row],h[col])@w1 + b1) @ w2 ----------
// h: [Nn,128], w1: [256,128], b1:[128], w2:[128]. K=256, N=128. WMMA tile + atomic dot.
__global__ __launch_bounds__(256)
void edgepred_wmma(const float* __restrict__ h, const int* __restrict__ row,
                   const int* __restrict__ col, const float* __restrict__ w1,
                   const float* __restrict__ b1, const float* __restrict__ w2,
                   float* __restrict__ ew, int E) {
  __shared__ float As[128][68];
  __shared__ float Bs[16][68];
  const int tid = threadIdx.x;
  const int lane = tid & 31;
  const int wave = tid >> 5;
  const int m0 = blockIdx.y * 128;
  const int n0 = blockIdx.x * 16;
  const int mloc = lane & 15;
  const int khalf = (lane >> 4) << 1;
  const int N = 128;

  v8f acc = {0.f, 0.f, 0.f, 0.f, 0.f, 0.f, 0.f, 0.f};
  for (int k0 = 0; k0 < 256; k0 += 64) {
    const int* idx = (k0 < 128) ? row : col;  // concat halves
    int coffb = k0 & 64;
#pragma unroll
    for (int i = 0; i < 8; ++i) {
      int chunk = tid + 256 * i;
      int r = chunk >> 4;
      int c4 = (chunk & 15) << 2;
      int e = m0 + r;
      float4 v = make_float4(0.f, 0.f, 0.f, 0.f);
      if (e < E) {
        int node = idx[e];
        v = *(const float4*)(h + (size_t)node * 128 + coffb + c4);
      }
      *(float4*)&As[r][c4] = v;
    }
    {
      int n = tid & 15;
      int kb = tid >> 4;
#pragma unroll
      for (int i = 0; i < 4; ++i) {
        int k = kb + 16 * i;
        Bs[n][k] = w1[(size_t)(k0 + k) * N + n0 + n];
      }
    }
    __syncthreads();
#pragma unroll
    for (int kk = 0; kk < 64; kk += 4) {
      v2f a = *(const v2f*)&As[wave * 16 + mloc][kk + khalf];
      v2f b = *(const v2f*)&Bs[mloc][kk + khalf];
      acc = __builtin_amdgcn_wmma_f32_16x16x4_f32(false, a, false, b,
                                                  (short)0, acc, false, false);
    }
    __syncthreads();
  }
  const int cn = n0 + (lane & 15);
  const float w2n = w2[cn];
  const float b1n = b1[cn];
  const int rbase = m0 + wave * 16 + ((lane >> 4) << 3);
#pragma unroll
  for (int j = 0; j < 8; ++j) {
    int e = rbase + j;
    if (e < E) {
      float z = fmaxf(acc[j] + b1n, 0.f);
      atomicAdd(&ew[e], z * w2n);  // partial dot over this lane's column
    }
  }
}

// =====================================================================
extern "C" void kernel_launch(void* const* d_in, const int* in_sizes, int n_in,
                              void* d_out, int out_size, void* d_ws, size_t ws_size,
                              hipStream_t stream) {
  (void)n_in; (void)out_size; (void)ws_size;
  const float* x     = (const float*)d_in[0];
  const int*   ei    = (const int*)d_in[1];
  const float* W0    = (const float*)d_in[2];
  const float* as0   = (const float*)d_in[3];
  const float* ad0   = (const float*)d_in[4];
  const float* b0    = (const float*)d_in[5];
  const float* Wm    = (const float*)d_in[6];
  const float* asm_p = (const float*)d_in[7];
  const float* adm_p = (const float*)d_in[8];
  const float* bm    = (const float*)d_in[9];
  const float* Wl    = (const float*)d_in[10];
  const float* asl   = (const float*)d_in[11];
  const float* adl   = (const float*)d_in[12];
  const float* bl    = (const float*)d_in[13];
  const float* ln_g  = (const float*)d_in[14];
  const float* ln_b  = (const float*)d_in[15];
  const float* ln_gl = (const float*)d_in[16];
  const float* ln_bl = (const float*)d_in[17];
  const float* ep_w1 = (const float*)d_in[18];
  const float* ep_b1 = (const float*)d_in[19];
  const float* ep_w2 = (const float*)d_in[20];
  const float* ep_b2 = (const float*)d_in[21];
  float* ew = (float*)d_out;

  const int Nn = in_sizes[0] / 128;
  const int E  = in_sizes[1] / 2;
  const int ET = E + Nn;
  const int* row = ei;
  const int* col = ei + E;

  float* ws = (float*)d_ws;
  size_t off = 0;
  auto take = [&](size_t nf) -> float* {
    float* p = ws + off;
    off += (nf + 63) & ~(size_t)63;
    return p;
  };
  float* W1buf = take((size_t)Nn * 512);
  float* W2buf = take((size_t)Nn * 512);
  float* Zbuf  = take((size_t)Nn * 512);
  float* alpS  = take((size_t)Nn * 4);
  float* alpD  = take((size_t)Nn * 4);
  unsigned int* segmax = (unsigned int*)take((size_t)Nn * 4);
  float* denom = take((size_t)Nn * 4);
  float* ebuf  = take((size_t)ET * 4);

  auto zero = [&](float* p, size_t nf) {
    fill_zero_f<<<(unsigned)((nf + 255) / 256), 256, 0, stream>>>(p, nf);
  };

  auto gat = [&](const float* hin, int Cin, const float* W, const float* a_s,
                 const float* a_d, const float* bias, const float* g,
                 const float* b, const float* resid, float* xp, float* out,
                 int H, int C) {
    int Cout = H * C;
    dim3 gg(Cout / 16, (Nn + 127) / 128);
    gemm_wmma_f32<<<gg, 256, 0, stream>>>(hin, W, xp, Nn, Cin, Cout);
    int nw = Nn * H;
    alpha_dots<<<(nw * 32 + 255) / 256, 256, 0, stream>>>(xp, a_s, a_d, alpS,
                                                          alpD, Nn, H, C);
    zero((float*)segmax, (size_t)Nn * H);
    zero(denom, (size_t)Nn * H);
    zero(Zbuf, (size_t)Nn * Cout);
    int gb = (ET + 255) / 256;
    edge_scores_max<<<gb, 256, 0, stream>>>(row, col, alpS, alpD, ebuf, segmax,
                                            E, Nn, H);
    edge_exp_sum<<<gb, 256, 0, stream>>>(row, col, ebuf, segmax, denom, E, Nn, H);
    edge_aggregate<<<ET, 256, 0, stream>>>(row, col, ebuf, denom, xp, Zbuf, E,
                                           Nn, H, C);
    finalize_ln_elu<<<Nn, 256, 0, stream>>>(Zbuf, bias, g, b, resid, out, Cout);
  };

  // layer 0: 128 -> 512 (no residual), h -> W2buf
  gat(x, 128, W0, as0, ad0, b0, ln_g, ln_b, nullptr, W1buf, W2buf, 4, 128);
  // layers 1..3: 512 -> 512 with residual (ping-pong W2/W1)
  gat(W2buf, 512, Wm + 0 * 512 * 512, asm_p + 0 * 512, adm_p + 0 * 512,
      bm + 0 * 512, ln_g + 1 * 512, ln_b + 1 * 512, W2buf, W1buf, W1buf, 4, 128);
  gat(W1buf, 512, Wm + 1 * 512 * 512, asm_p + 1 * 512, adm_p + 1 * 512,
      bm + 1 * 512, ln_g + 2 * 512, ln_b + 2 * 512, W1buf, W2buf, W2buf, 4, 128);
  gat(W2buf, 512, Wm + 2 * 512 * 512, asm_p + 2 * 512, adm_p + 2 * 512,
      bm + 2 * 512, ln_g + 3 * 512, ln_b + 3 * 512, W2buf, W1buf, W1buf, 4, 128);
  // last layer: 512 -> 128, H=1 (mean over 1 head == identity), no residual
  gat(W1buf, 512, Wl, asl, adl, bl, ln_gl, ln_bl, nullptr, W2buf, W2buf, 1, 128);

  // edge predictor on original edges; h_final = W2buf ([Nn,128])
  init_ew<<<(E + 255) / 256, 256, 0, stream>>>(ew, ep_b2, E);
  dim3 gp(128 / 16, (E + 127) / 128);
  edgepred_wmma<<<gp, 256, 0, stream>>>(W2buf, row, col, ep_w1, ep_b1, ep_w2,
                                        ew, E);
}